// AttentionBlock_79413945303525
// MI455X (gfx1250) — compile-verified
//
#include <hip/hip_runtime.h>
#include <hip/hip_bf16.h>
#include <math.h>

#define B_  16
#define SQ_ 1024
#define SK_ 1024
#define H_  256
#define NH_ 8
#define HD_ 32

typedef __attribute__((ext_vector_type(2))) float v2f;
typedef __attribute__((ext_vector_type(8))) float v8f;

// D = A(16x4) * B(4x16) + C(16x16), fp32, wave32.  (codegen-confirmed)
__device__ __forceinline__ v8f wmma_f32(v2f a, v2f b, v8f c) {
  return __builtin_amdgcn_wmma_f32_16x16x4_f32(false, a, false, b, (short)0, c,
                                               false, false);
}

// -------------------------------------------------------------------------
// C[M,N] = A[M,K] * W[N,K]^T   (i.e. x @ W.T), all row-major fp32.
// One wave -> one 16x16 C tile. A-frag lane L: float2 A[m0+(L&15)][k+2*(L>>4)]
// B-frag lane L: float2 W[n0+(L&15)][k+2*(L>>4)]  (B[k',n] = W[n,k'])
// -------------------------------------------------------------------------
__global__ __launch_bounds__(256) void proj_gemm_kernel(
    const float* __restrict__ A, const float* __restrict__ W,
    float* __restrict__ C, int M, int N, int K) {
  const int lane = threadIdx.x & 31;
  const int wave = threadIdx.x >> 5;
  const int l16  = lane & 15;
  const int half = lane >> 4;
  const int m0 = blockIdx.y * 128 + wave * 16;
  const int n0 = blockIdx.x * 16;

  const float* arow = A + (size_t)(m0 + l16) * K + 2 * half;
  const float* wrow = W + (size_t)(n0 + l16) * K + 2 * half;

  v8f acc = {};
  for (int k = 0; k < K; k += 4) {
    v2f av = *(const v2f*)(arow + k);
    v2f bv = *(const v2f*)(wrow + k);
    acc = wmma_f32(av, bv, acc);
  }
  // C layout: VGPR r -> row m0 + r + 8*half, col n0 + l16
  float* crow = C + (size_t)(m0 + 8 * half) * N + n0 + l16;
#pragma unroll
  for (int r = 0; r < 8; ++r) crow[(size_t)r * N] = acc[r];
}

// -------------------------------------------------------------------------
// Attention: per block = (b, nh, 128 q-rows), 8 waves x 16 q-rows.
// Pass A: LANE-LOCAL online softmax stats (each lane only ever sees column
//         kt+l16 of each tile), merged across 16 lanes once at the end via
//         log-sum-exp -> no cross-lane traffic in the streaming loop.
// Pass B: recompute S, write normalized attention (nontemporal) and
//         accumulate column sums (for xs = colsum(P) . V) into LDS -> global.
// -------------------------------------------------------------------------
__global__ __launch_bounds__(256) void attn_kernel(
    const float* __restrict__ Qp, const float* __restrict__ Kp,
    const float* __restrict__ bias, float* __restrict__ attn,
    float* __restrict__ colsum) {
  __shared__ float cs_s[SK_];
  const int b  = blockIdx.x;   // innermost: 16 batches share bias slab in L2
  const int qb = blockIdx.y;
  const int nh = blockIdx.z;
  const int tid  = threadIdx.x;
  const int lane = tid & 31;
  const int wave = tid >> 5;
  const int l16  = lane & 15;
  const int half = lane >> 4;

  for (int kk = tid; kk < SK_; kk += 256) cs_s[kk] = 0.f;
  __syncthreads();

  const int q0 = qb * 128 + wave * 16;
  const float scale = 0.17677669529663687f;  // 1/sqrt(32)

  // Q A-frags for the whole 16x32 Q tile (8 WMMA steps over HD)
  const float* qrow =
      Qp + ((size_t)(b * SQ_ + q0 + l16)) * H_ + nh * HD_ + 2 * half;
  v2f qa[8];
#pragma unroll
  for (int i = 0; i < 8; ++i) qa[i] = *(const v2f*)(qrow + 4 * i);

  const float* biasb =
      bias + ((size_t)nh * SQ_ + q0 + 8 * half) * SK_ + l16;

  // Per-lane running stats over the column subset this lane observes.
  float m[8], l[8];
#pragma unroll
  for (int r = 0; r < 8; ++r) { m[r] = -1e30f; l[r] = 0.f; }

  // ---- Pass A: lane-local online max / sum-exp (no shuffles inside) ----
  for (int kt = 0; kt < SK_; kt += 16) {
    const float* krow =
        Kp + ((size_t)(b * SK_ + kt + l16)) * H_ + nh * HD_ + 2 * half;
    v8f s = {};
#pragma unroll
    for (int i = 0; i < 8; ++i) {
      v2f kv = *(const v2f*)(krow + 4 * i);  // B[d,k]=Kp[k,d]: contiguous
      s = wmma_f32(qa[i], kv, s);
    }
#pragma unroll
    for (int r = 0; r < 8; ++r) {
      float e = s[r] * scale + biasb[(size_t)r * SK_ + kt];
      float nm = fmaxf(m[r], e);
      l[r] = l[r] * __expf(m[r] - nm) + __expf(e - nm);
      m[r] = nm;
    }
  }

  // Merge the 16 lane-local (m,l) pairs per row via log-sum-exp (once).
  float il[8];
#pragma unroll
  for (int r = 0; r < 8; ++r) {
    float mm = m[r], ll = l[r];
#pragma unroll
    for (int mask = 1; mask <= 8; mask <<= 1) {
      float mo = __shfl_xor(mm, mask);
      float lo = __shfl_xor(ll, mask);
      float nm = fmaxf(mm, mo);
      ll = ll * __expf(mm - nm) + lo * __expf(mo - nm);
      mm = nm;
    }
    m[r]  = mm;
    il[r] = 1.0f / ll;
  }

  // ---- Pass B: write attention + column sums ----
  float* attb =
      attn + (((size_t)b * NH_ + nh) * SQ_ + q0 + 8 * half) * SK_ + l16;
  for (int kt = 0; kt < SK_; kt += 16) {
    const float* krow =
        Kp + ((size_t)(b * SK_ + kt + l16)) * H_ + nh * HD_ + 2 * half;
    v8f s = {};
#pragma unroll
    for (int i = 0; i < 8; ++i) {
      v2f kv = *(const v2f*)(krow + 4 * i);
      s = wmma_f32(qa[i], kv, s);
    }
    float csum = 0.f;
#pragma unroll
    for (int r = 0; r < 8; ++r) {
      float e = s[r] * scale + biasb[(size_t)r * SK_ + kt];
      float p = __expf(e - m[r]) * il[r];
      // 512 MiB, never re-read: bypass cache retention
      __builtin_nontemporal_store(p, attb + (size_t)r * SK_ + kt);
      csum += p;
    }
    csum += __shfl_xor(csum, 16);  // combine rows 0-7 with rows 8-15
    if (half == 0) atomicAdd(&cs_s[kt + l16], csum);
  }
  __syncthreads();
  float* csg = colsum + ((size_t)b * NH_ + nh) * SK_;
  for (int kk = tid; kk < SK_; kk += 256) atomicAdd(&csg[kk], cs_s[kk]);
}

// -------------------------------------------------------------------------
// Epilogue per batch: res = sum_q q ; xs = colsum(P) . V ;
// h = xs@W1^T + b1 ; BN(eval) ; mish ; + res
// -------------------------------------------------------------------------
__global__ __launch_bounds__(256) void final_kernel(
    const float* __restrict__ qin, const float* __restrict__ Vp,
    const float* __restrict__ colsum, const float* __restrict__ W1,
    const float* __restrict__ b1, const float* __restrict__ gamma,
    const float* __restrict__ beta, const float* __restrict__ rmean,
    const float* __restrict__ rvar, float* __restrict__ out) {
  __shared__ float xs_s[H_];
  const int b = blockIdx.x;
  const int t = threadIdx.x;  // one output channel per thread

  float resv = 0.f;
  const float* qb = qin + (size_t)b * SQ_ * H_ + t;
  for (int s = 0; s < SQ_; ++s) resv += qb[(size_t)s * H_];

  const int nh = t / HD_;
  const float* cs = colsum + ((size_t)b * NH_ + nh) * SK_;
  const float* vp = Vp + (size_t)b * SK_ * H_ + t;  // t = nh*HD + d
  float xsv = 0.f;
  for (int k = 0; k < SK_; ++k) xsv += cs[k] * vp[(size_t)k * H_];
  xs_s[t] = xsv;
  __syncthreads();

  float hv = b1[t];
  const float* w1r = W1 + (size_t)t * H_;
  for (int k = 0; k < H_; ++k) hv += xs_s[k] * w1r[k];

  float hn = (hv - rmean[t]) * rsqrtf(rvar[t] + 1e-5f) * gamma[t] + beta[t];
  float sp = (hn > 20.f) ? hn : log1pf(__expf(hn));  // stable softplus
  out[(size_t)b * H_ + t] = hn * tanhf(sp) + resv;
}

extern "C" void kernel_launch(void* const* d_in, const int* in_sizes, int n_in,
                              void* d_out, int out_size, void* d_ws,
                              size_t ws_size, hipStream_t stream) {
  const float* q     = (const float*)d_in[0];
  const float* k     = (const float*)d_in[1];
  const float* v     = (const float*)d_in[2];
  const float* bias  = (const float*)d_in[3];
  const float* W_Q   = (const float*)d_in[4];
  const float* W_K   = (const float*)d_in[5];
  const float* W_V   = (const float*)d_in[6];
  const float* W1    = (const float*)d_in[7];
  const float* b1    = (const float*)d_in[8];
  const float* gamma = (const float*)d_in[9];
  const float* beta  = (const float*)d_in[10];
  const float* rmean = (const float*)d_in[11];
  const float* rvar  = (const float*)d_in[12];

  float* out  = (float*)d_out;                 // [B,H] first
  float* attn = out + (size_t)B_ * H_;         // then [B,NH,SQ,SK]

  char* ws = (char*)d_ws;
  const size_t projBytes = (size_t)B_ * SQ_ * H_ * sizeof(float);  // 16 MiB
  float* Qp     = (float*)(ws);
  float* Kp     = (float*)(ws + projBytes);
  float* Vp     = (float*)(ws + 2 * projBytes);
  float* colsum = (float*)(ws + 3 * projBytes);
  const size_t colBytes = (size_t)B_ * NH_ * SK_ * sizeof(float);

  hipMemsetAsync(colsum, 0, colBytes, stream);

  dim3 pg(H_ / 16, (B_ * SQ_) / 128);
  proj_gemm_kernel<<<pg, 256, 0, stream>>>(q, W_Q, Qp, B_ * SQ_, H_, H_);
  proj_gemm_kernel<<<pg, 256, 0, stream>>>(k, W_K, Kp, B_ * SK_, H_, H_);
  proj_gemm_kernel<<<pg, 256, 0, stream>>>(v, W_V, Vp, B_ * SK_, H_, H_);

  attn_kernel<<<dim3(B_, SQ_ / 128, NH_), 256, 0, stream>>>(Qp, Kp, bias, attn,
                                                            colsum);

  final_kernel<<<dim3(B_), dim3(H_), 0, stream>>>(q, Vp, colsum, W1, b1, gamma,
                                                  beta, rmean, rvar, out);
}